// LSTMModel_1872605741073
// MI455X (gfx1250) — compile-verified
//
#include <hip/hip_runtime.h>
#include <hip/hip_bf16.h>
#include <math.h>

// MI455X (gfx1250) 3-layer LSTM, fp32, batch-parallel recurrence.
// Each block owns 16 batch rows for all 512 time steps; combined [Wih;Whh]
// lives in (dynamic) LDS in WMMA B-operand order for the whole scan.
// Per-step GEMM uses v_wmma_f32_16x16x4_f32 (fp32-exact vs reference).
// Gate nonlinearities use hardware v_tanh_f32 (sigmoid via exact identity).
// Workspace needed: 2 * 512*512*100 * 4 B = 209,715,200 bytes.

typedef float v2f __attribute__((ext_vector_type(2)));
typedef float v8f __attribute__((ext_vector_type(8)));

#define Hh   100          // hidden size
#define G4   400          // 4 * H (gate columns)
#define Bsz  512
#define Tsz  512
#define ROWS 16           // batch rows per block
#define NWAVES 7          // ceil(100/16) hidden tiles; wave w owns cols [16w,16w+16)
#define NTHREADS (NWAVES * 32)

// Hardware TRANS tanh. ISA 7.x: one independent op (or V_NOP) required after a
// TRANS op before its result may be consumed -- enforced inside the asm since
// the compiler cannot model the hazard through inline asm.
__device__ __forceinline__ float tanh_hw(float x) {
    float r;
    asm("v_tanh_f32 %0, %1\n\tv_nop" : "=v"(r) : "v"(x));
    return r;
}
// Exact: sigmoid(x) = 0.5 * tanh(0.5 x) + 0.5
__device__ __forceinline__ float sig_hw(float x) {
    return fmaf(0.5f, tanh_hw(0.5f * x), 0.5f);
}

// x: [B,T,I]  Wih: [400,I]  Whh: [400,100]  bih,bhh: [400]  hout: [B,T,100]
template<int I, int IP>   // IP = I padded to multiple of 4 (zero-filled weight rows)
__global__ __launch_bounds__(NTHREADS)
void lstm_layer_kernel(const float* __restrict__ x,
                       const float* __restrict__ Wih,
                       const float* __restrict__ Whh,
                       const float* __restrict__ bih,
                       const float* __restrict__ bhh,
                       float* __restrict__ hout)
{
    constexpr int KT = IP + Hh;            // combined K (multiple of 4)
    extern __shared__ char smem_raw[];
    v2f*   Wlds = (v2f*)smem_raw;                                        // [(KT/2)*400]
    float* hlds = (float*)(smem_raw + (size_t)(KT/2) * G4 * sizeof(v2f)); // [16*100]

    const int tid  = threadIdx.x;
    const int wave = tid >> 5;             // 0..6
    const int lane = tid & 31;
    const int ln   = lane & 15;            // tile row/col index
    const int lg   = lane >> 4;            // K-pair / M-half selector
    const int b0   = blockIdx.x * ROWS;

    // ---- one-time: combined weight matrix -> LDS, B-operand order ----
    // Wlds[kk*400 + n] = { Wt[2kk][n], Wt[2kk+1][n] } where Wt[k][n] is
    // Wih[n][k] for k<IP (0-padded past I) and Whh[n][k-IP] otherwise.
    for (int idx = tid; idx < (KT/2) * G4; idx += NTHREADS) {
        int kk = idx / G4, n = idx - kk * G4;
        int k  = kk * 2;
        v2f v;
        if (k < IP) {
            v.x = (k     < I) ? Wih[n * I + k    ] : 0.f;
            v.y = (k + 1 < I) ? Wih[n * I + k + 1] : 0.f;
        } else {
            int kh = k - IP;
            v.x = Whh[n * Hh + kh];
            v.y = Whh[n * Hh + kh + 1];
        }
        Wlds[idx] = v;
    }
    for (int idx = tid; idx < ROWS * Hh; idx += NTHREADS) hlds[idx] = 0.f;
    __syncthreads();

    // per-lane gate biases: this lane's hidden column j = 16*wave + ln
    const int j  = 16 * wave + ln;
    const int jc = (j < Hh) ? j : (Hh - 1);      // clamp for the ragged 7th wave
    float bias[4];
    #pragma unroll
    for (int g = 0; g < 4; ++g) bias[g] = bih[g * Hh + jc] + bhh[g * Hh + jc];

    // cell state stays in registers: c[m][j] for m = r + 8*lg
    float cReg[8];
    #pragma unroll
    for (int r = 0; r < 8; ++r) cReg[r] = 0.f;

    const float* xrow0 = x + (size_t)(b0 + ln) * Tsz * I;   // row m = ln, t = 0

    for (int t = 0; t < Tsz; ++t) {
        v8f acc[4];                              // gates i,f,g,o (D-tile layout)
        #pragma unroll
        for (int g = 0; g < 4; ++g) {
            #pragma unroll
            for (int r = 0; r < 8; ++r) acc[g][r] = bias[g];
        }

        const float* xrow = xrow0 + (size_t)t * I;
        if (t + 1 < Tsz) __builtin_prefetch(xrow + I, 0, 1);  // next step's x row

        // ---- x contribution: A operands straight from global (L2/WGP$-hot) ----
        // For k >= I the LDS weight rows are zero, so a clamped (unguarded)
        // load is exact: garbage * 0 == 0. No exec-mask branches.
        #pragma unroll 5
        for (int kt = 0; kt < IP / 4; ++kt) {
            int k = kt * 4 + 2 * lg;
            v2f a = *(const v2f*)(xrow + ((k < I) ? k : 0));
            const v2f* wrow = Wlds + (size_t)(kt * 2 + lg) * G4;
            #pragma unroll
            for (int g = 0; g < 4; ++g) {
                v2f bb = wrow[g * Hh + jc];
                acc[g] = __builtin_amdgcn_wmma_f32_16x16x4_f32(
                    false, a, false, bb, (short)0, acc[g], false, false);
            }
        }

        // ---- h contribution: A operands from LDS h buffer ----
        #pragma unroll 5
        for (int kt = 0; kt < Hh / 4; ++kt) {
            int k = kt * 4 + 2 * lg;
            v2f a = *(const v2f*)&hlds[ln * Hh + k];
            const v2f* wrow = Wlds + (size_t)(IP / 2 + kt * 2 + lg) * G4;
            #pragma unroll
            for (int g = 0; g < 4; ++g) {
                v2f bb = wrow[g * Hh + jc];
                acc[g] = __builtin_amdgcn_wmma_f32_16x16x4_f32(
                    false, a, false, bb, (short)0, acc[g], false, false);
            }
        }

        __syncthreads();                 // all waves done reading h_{t-1}

        const bool active = (j < Hh);
        #pragma unroll
        for (int r = 0; r < 8; ++r) {
            float iv = sig_hw(acc[0][r]);
            float fv = sig_hw(acc[1][r]);
            float gv = tanh_hw(acc[2][r]);
            float ov = sig_hw(acc[3][r]);
            float cv = fmaf(fv, cReg[r], iv * gv);
            cReg[r]  = cv;
            float hv = ov * tanh_hw(cv);
            int   m  = r + 8 * lg;       // D-tile row
            if (active) {
                hlds[m * Hh + j] = hv;
                hout[((size_t)(b0 + m) * Tsz + t) * Hh + j] = hv;
            }
        }
        __syncthreads();                 // h_t visible before next step's reads
    }
}

__global__ void dense_kernel(const float* __restrict__ h3,
                             const float* __restrict__ Wd,
                             const float* __restrict__ bd,
                             float* __restrict__ out)
{
    int b = blockIdx.x * blockDim.x + threadIdx.x;
    if (b >= Bsz) return;
    const float* row = h3 + ((size_t)b * Tsz + (Tsz - 1)) * Hh;
    float s = 0.f;
    #pragma unroll 4
    for (int k = 0; k < Hh; ++k) s += row[k] * Wd[k];
    out[b] = s + bd[0];
}

extern "C" void kernel_launch(void* const* d_in, const int* in_sizes, int n_in,
                              void* d_out, int out_size, void* d_ws, size_t ws_size,
                              hipStream_t stream) {
    const float* input = (const float*)d_in[0];   // [512,512,10]
    const float* Wih1  = (const float*)d_in[1];
    const float* Whh1  = (const float*)d_in[2];
    const float* bih1  = (const float*)d_in[3];
    const float* bhh1  = (const float*)d_in[4];
    const float* Wih2  = (const float*)d_in[5];
    const float* Whh2  = (const float*)d_in[6];
    const float* bih2  = (const float*)d_in[7];
    const float* bhh2  = (const float*)d_in[8];
    const float* Wih3  = (const float*)d_in[9];
    const float* Whh3  = (const float*)d_in[10];
    const float* bih3  = (const float*)d_in[11];
    const float* bhh3  = (const float*)d_in[12];
    const float* Wd    = (const float*)d_in[13];
    const float* bd    = (const float*)d_in[14];
    float* out = (float*)d_out;

    float* hbuf0 = (float*)d_ws;                              // [512,512,100]
    float* hbuf1 = hbuf0 + (size_t)Bsz * Tsz * Hh;            // [512,512,100]

    const size_t shm1  = (size_t)((12 + Hh) / 2) * G4 * sizeof(v2f) + ROWS * Hh * sizeof(float);
    const size_t shm23 = (size_t)((Hh + Hh) / 2) * G4 * sizeof(v2f) + ROWS * Hh * sizeof(float);

    dim3 grid(Bsz / ROWS), block(NTHREADS);
    lstm_layer_kernel<10, 12><<<grid, block, shm1,  stream>>>(input, Wih1, Whh1, bih1, bhh1, hbuf0);
    lstm_layer_kernel<Hh, Hh><<<grid, block, shm23, stream>>>(hbuf0, Wih2, Whh2, bih2, bhh2, hbuf1);
    lstm_layer_kernel<Hh, Hh><<<grid, block, shm23, stream>>>(hbuf1, Wih3, Whh3, bih3, bhh3, hbuf0);
    dense_kernel<<<dim3(2), dim3(256), 0, stream>>>(hbuf0, Wd, bd, out);
}